// MyModel_61933428416168
// MI455X (gfx1250) — compile-verified
//
#include <hip/hip_runtime.h>

// Problem constants (from reference: x is (64, 1024, 1024) f32)
#define NBATCH   64
#define NDIM     1024
#define TS       64                     // tile side
#define TPD      (NDIM / TS)            // 16 tiles per dimension
#define NPAIRS   (TPD * (TPD + 1) / 2)  // 136 unordered tile pairs (I<=J)
#define LDSTRIDE 68                     // 64 + 4 pad: keeps rows 16B-aligned, breaks bank repeats
#define THREADS  256

#if defined(__gfx1250__) && __has_builtin(__builtin_amdgcn_global_load_async_to_lds_b128)
#define USE_ASYNC_LDS 1
#else
#define USE_ASYNC_LDS 0
#endif

// Exact parameter types for the async builtin: <4 x i32> pointers in AS1 / AS3.
typedef int v4i __attribute__((__vector_size__(4 * sizeof(int))));
typedef __attribute__((address_space(1))) v4i g_v4i;
typedef __attribute__((address_space(3))) v4i l_v4i;

__global__ void trace_init_kernel(float* __restrict__ out) {
    int t = threadIdx.x;
    if (t < 2 * NBATCH) out[t] = 0.0f;
}

__global__ __launch_bounds__(THREADS)
void trace_pair_kernel(const float* __restrict__ x, float* __restrict__ out) {
    __shared__ float sA[TS * LDSTRIDE];
    __shared__ float sB[TS * LDSTRIDE];
    __shared__ float red2[THREADS / 32];
    __shared__ float red1[THREADS / 32];

    const int tid = threadIdx.x;
    int p       = (int)blockIdx.x % NPAIRS;
    const int b = (int)blockIdx.x / NPAIRS;

    // decode triangular pair index -> (I, J), I <= J   (scalar, SALU-only)
    int I = 0;
    while (p >= TPD - I) { p -= TPD - I; ++I; }
    const int J     = I + p;
    const bool diag = (I == J);

    const float* baseA = x + (size_t)b * NDIM * NDIM + (size_t)(I * TS) * NDIM + (size_t)(J * TS);
    const float* baseB = x + (size_t)b * NDIM * NDIM + (size_t)(J * TS) * NDIM + (size_t)(I * TS);

    // ---- Stage tile(s) into LDS: 4096 floats = 1024 x 16B chunks, 4 chunks/thread ----
#if USE_ASYNC_LDS
    {
#pragma unroll
        for (int k = 0; k < 4; ++k) {
            const int c  = tid + k * THREADS;   // chunk id 0..1023
            const int r  = c >> 4;              // row 0..63
            const int cc = (c & 15) << 2;       // float offset within row (16B chunks)
            __builtin_amdgcn_global_load_async_to_lds_b128(
                (g_v4i*)(baseA + (size_t)r * NDIM + cc),
                (l_v4i*)(&sA[r * LDSTRIDE + cc]), 0, 0);
            if (!diag) {
                __builtin_amdgcn_global_load_async_to_lds_b128(
                    (g_v4i*)(baseB + (size_t)r * NDIM + cc),
                    (l_v4i*)(&sB[r * LDSTRIDE + cc]), 0, 0);
            }
        }
        __builtin_amdgcn_s_wait_asynccnt(0);
    }
#else
    {
#pragma unroll
        for (int k = 0; k < 4; ++k) {
            const int c  = tid + k * THREADS;
            const int r  = c >> 4;
            const int cc = (c & 15) << 2;
            const float4 va = *(const float4*)(baseA + (size_t)r * NDIM + cc);
            *(float4*)(&sA[r * LDSTRIDE + cc]) = va;
            if (!diag) {
                const float4 vb = *(const float4*)(baseB + (size_t)r * NDIM + cc);
                *(float4*)(&sB[r * LDSTRIDE + cc]) = vb;
            }
        }
    }
#endif
    __syncthreads();

    // ---- Compute: acc2 = sum_{i,j in tile} A[i][j] * B[j][i];  acc1 = diag trace ----
    const float* TB = diag ? sA : sB;
    float acc2 = 0.0f;
    float acc1 = 0.0f;
#pragma unroll
    for (int k = 0; k < 16; ++k) {
        const int e = tid + k * THREADS;   // element 0..4095
        const int i = e >> 6;
        const int j = e & 63;
        const float a  = sA[i * LDSTRIDE + j];   // row read: conflict-free
        const float bt = TB[j * LDSTRIDE + i];   // col read: 2-way conflict in wave32
        acc2 += a * bt;
        if (diag && i == j) acc1 += a;
    }
    if (!diag) acc2 *= 2.0f;   // block (J,I) contributes identically

    // ---- Reduce within wave32, then across 8 waves, one atomic per block ----
#pragma unroll
    for (int off = 16; off > 0; off >>= 1) {
        acc2 += __shfl_down(acc2, off, 32);
        acc1 += __shfl_down(acc1, off, 32);
    }
    const int wid = tid >> 5, lane = tid & 31;
    if (lane == 0) { red2[wid] = acc2; red1[wid] = acc1; }
    __syncthreads();
    if (tid == 0) {
        float s2 = 0.0f, s1 = 0.0f;
#pragma unroll
        for (int w = 0; w < THREADS / 32; ++w) { s2 += red2[w]; s1 += red1[w]; }
        atomicAdd(&out[NBATCH + b], s2);       // trace(x @ x)
        if (diag) atomicAdd(&out[b], s1);      // trace(x)
    }
}

extern "C" void kernel_launch(void* const* d_in, const int* in_sizes, int n_in,
                              void* d_out, int out_size, void* d_ws, size_t ws_size,
                              hipStream_t stream) {
    (void)in_sizes; (void)n_in; (void)d_ws; (void)ws_size; (void)out_size;
    const float* x = (const float*)d_in[0];
    float* out     = (float*)d_out;

    trace_init_kernel<<<1, 128, 0, stream>>>(out);
    trace_pair_kernel<<<NBATCH * NPAIRS, THREADS, 0, stream>>>(x, out);
}